// DNC_46943992545706
// MI455X (gfx1250) — compile-verified
//
#include <hip/hip_runtime.h>
#include <math.h>

#define B_ 16
#define T_ 32
#define W_ 128
#define R_ 4
#define N_ 512
#define M_ 128
#define DIN 768          // W + R*W + W
#define XI_ 919          // W*R + 3*W + 5*R + 3
#define EPS_ 1e-6f

typedef float v2f __attribute__((ext_vector_type(2)));
typedef float v8f __attribute__((ext_vector_type(8)));
typedef unsigned int u32x4 __attribute__((ext_vector_type(4)));
typedef int i32x4 __attribute__((ext_vector_type(4)));
typedef int i32x8 __attribute__((ext_vector_type(8)));

#if defined(__has_builtin)
#if __has_builtin(__builtin_amdgcn_tensor_load_to_lds)
#define HAVE_TDM 1
#endif
#endif

__device__ __forceinline__ float sigm(float x) { return 1.0f / (1.0f + expf(-x)); }
__device__ __forceinline__ float oneplus(float x) { return 1.0f + log1pf(expf(x) + EPS_); }

// D = A(16x4,f32) * B(4x16,f32) + C ; full-precision WMMA (VOP3P op 93)
__device__ __forceinline__ v8f wmma4(v2f a, v2f b, v8f c) {
  return __builtin_amdgcn_wmma_f32_16x16x4_f32(false, a, false, b, (short)0, c, false, false);
}

#ifdef HAVE_TDM
// 2D tile (tile_d0 elems/row x tile_d1 rows, f32) from global -> LDS via Tensor Data Mover.
// D# bit layout per CDNA5 ISA ch.8: group0 {count, lds_addr, global_addr, type=2},
// group1 {data_size=4B, tensor dims, tile dims, dim0 stride}. Remaining groups zero (2D tile).
// This toolchain exposes the 6-arg builtin (g0, g1, g2, g3, g4, cpol).
__device__ __forceinline__ void tdm_load_2d(unsigned lds_off, const void* gp,
                                            unsigned tile_d0, unsigned tile_d1,
                                            unsigned tensor_d0, unsigned tensor_d1,
                                            unsigned long long stride0_elems) {
  unsigned long long ga = (unsigned long long)gp;
  u32x4 g0;
  g0[0] = 1u;                                             // count=1 (valid user D#)
  g0[1] = lds_off;                                        // lds_addr (bytes)
  g0[2] = (unsigned)(ga & 0xffffffffu);                   // global_addr[31:0]
  g0[3] = (unsigned)((ga >> 32) & 0x01ffffffu) | (2u << 30); // addr[56:32] | type=2
  i32x8 g1;
  g1[0] = (int)(2u << 16);                                // data_size=2 (4 bytes)
  g1[1] = (int)((tensor_d0 & 0xffffu) << 16);             // tensor_dim0[15:0]
  g1[2] = (int)((tensor_d0 >> 16) | ((tensor_d1 & 0xffffu) << 16));
  g1[3] = (int)((tensor_d1 >> 16) | (tile_d0 << 16));     // tile_dim0
  g1[4] = (int)(tile_d1 & 0xffffu);                       // tile_dim1 (tile_dim2=0)
  g1[5] = (int)(stride0_elems & 0xffffffffu);             // tensor_dim0_stride lo
  g1[6] = (int)((stride0_elems >> 32) & 0xffffu);         // stride hi (dim1 stride=0)
  g1[7] = 0;
  i32x4 z4 = {0, 0, 0, 0};
  i32x8 z8 = {0, 0, 0, 0, 0, 0, 0, 0};
  __builtin_amdgcn_tensor_load_to_lds(g0, g1, z4, z4, z8, 0);
}
#endif

// ---------------------------------------------------------------- init
__global__ __launch_bounds__(256) void k_init_zero(float* p, unsigned n) {
  unsigned i = blockIdx.x * 256u + threadIdx.x;
  if (i < n) p[i] = 0.0f;
}
__global__ __launch_bounds__(256) void k_init_copy(const float* mem, const float* hs_in,
                                                   float* memv, float* hs) {
  unsigned i = blockIdx.x * 256u + threadIdx.x;
  if (i < (unsigned)(B_ * N_ * M_)) memv[i] = mem[i];
  if (i < (unsigned)(B_ * W_)) hs[i] = hs_in[i];
}

// ---------------------------------------------------------------- controller (WMMA + TDM)
__global__ __launch_bounds__(256)
void k_controller(const float* X, const int* reset,
                  const float* Wb, const float* bb,
                  const float* Wff1, const float* bff1,
                  const float* Wff2, const float* bff2,
                  const float* Wta, const float* bta,
                  const float* Wtb, const float* btb,
                  const float* Wxi, const float* bxi,
                  float* hs, const float* prev_reads, float* xi, int t) {
  __shared__ float Ax[B_][W_];        // x_t
  __shared__ float Ar[B_][R_ * W_];   // prev_reads
  __shared__ float Ah[B_][W_];        // hs*(1-r)
  __shared__ float Z[B_][W_];
  __shared__ float H[B_][W_];
  int tid = threadIdx.x;

#ifdef HAVE_TDM
  if (tid < 32) {   // TDM ignores EXEC; issue once from wave 0
    tdm_load_2d((unsigned)(unsigned long long)&Ax[0][0], X + (size_t)t * W_,
                W_, B_, W_, B_, (unsigned long long)T_ * W_);
    tdm_load_2d((unsigned)(unsigned long long)&Ar[0][0], prev_reads,
                R_ * W_, B_, R_ * W_, B_, (unsigned long long)R_ * W_);
    __builtin_amdgcn_s_wait_tensorcnt(0);
  }
#else
  for (int i = tid; i < B_ * W_; i += 256)
    Ax[i / W_][i % W_] = X[((i / W_) * T_ + t) * W_ + (i % W_)];
  for (int i = tid; i < B_ * R_ * W_; i += 256)
    Ar[i / (R_ * W_)][i % (R_ * W_)] = prev_reads[i];
#endif
  for (int i = tid; i < B_ * W_; i += 256) {
    int b = i / W_;
    float r = (float)reset[b * T_ + t];
    Ah[b][i % W_] = hs[i] * (1.0f - r);
  }
  __syncthreads();

  int wave = tid >> 5, lane = tid & 31;
  int arow = lane & 15;             // A row (batch) / B col within tile
  int ksel = (lane >> 4) << 1;      // 0 or 2 (K sub-select per ISA layout)
  int rbase = (lane >> 4) << 3;     // C/D rows: i or i+8

  { // ---- z = silu([x|reads|hs] @ Wb + bb), K segmented over the 3 LDS buffers
    int col = wave * 16 + arow;
    v8f acc = {};
    for (int k = 0; k < W_; k += 4) {
      v2f a, b;
      a.x = Ax[arow][k + ksel];            a.y = Ax[arow][k + ksel + 1];
      b.x = Wb[(k + ksel) * W_ + col];     b.y = Wb[(k + ksel + 1) * W_ + col];
      acc = wmma4(a, b, acc);
    }
    for (int k = 0; k < R_ * W_; k += 4) {
      v2f a, b;
      int kr = W_ + k;
      a.x = Ar[arow][k + ksel];            a.y = Ar[arow][k + ksel + 1];
      b.x = Wb[(kr + ksel) * W_ + col];    b.y = Wb[(kr + ksel + 1) * W_ + col];
      acc = wmma4(a, b, acc);
    }
    for (int k = 0; k < W_; k += 4) {
      v2f a, b;
      int kr = W_ + R_ * W_ + k;
      a.x = Ah[arow][k + ksel];            a.y = Ah[arow][k + ksel + 1];
      b.x = Wb[(kr + ksel) * W_ + col];    b.y = Wb[(kr + ksel + 1) * W_ + col];
      acc = wmma4(a, b, acc);
    }
    for (int i = 0; i < 8; ++i) {
      float v = acc[i] + bb[col];
      Z[rbase + i][col] = v * sigm(v);    // silu
    }
  }
  __syncthreads();

  { // ---- ff1/ff2/ta/tb -> hs
    int col = wave * 16 + arow;
    v8f c1 = {}, c2 = {}, c3 = {}, c4 = {};
    for (int k = 0; k < W_; k += 4) {
      v2f a, b;
      a.x = Z[arow][k + ksel]; a.y = Z[arow][k + ksel + 1];
      b.x = Wff1[(k + ksel) * W_ + col]; b.y = Wff1[(k + ksel + 1) * W_ + col];
      c1 = wmma4(a, b, c1);
      b.x = Wff2[(k + ksel) * W_ + col]; b.y = Wff2[(k + ksel + 1) * W_ + col];
      c2 = wmma4(a, b, c2);
      b.x = Wta[(k + ksel) * W_ + col];  b.y = Wta[(k + ksel + 1) * W_ + col];
      c3 = wmma4(a, b, c3);
      b.x = Wtb[(k + ksel) * W_ + col];  b.y = Wtb[(k + ksel + 1) * W_ + col];
      c4 = wmma4(a, b, c4);
    }
    float tf = (float)t;
    for (int i = 0; i < 8; ++i) {
      float f1 = c1[i] + bff1[col];
      float f2 = c2[i] + bff2[col];
      float ti = sigm((c3[i] + bta[col]) * tf + (c4[i] + btb[col]));
      H[rbase + i][col] = f1 * (1.0f - ti) + ti * f2;
    }
  }
  __syncthreads();
  for (int i = tid; i < B_ * W_; i += 256) hs[i] = H[i / W_][i % W_];

  // ---- xi = hs @ Wxi + bxi (58 column tiles, guarded at 919)
  for (int tile = wave; tile < (XI_ + 15) / 16; tile += 8) {
    int col = tile * 16 + arow;
    bool okc = col < XI_;
    v8f acc = {};
    for (int k = 0; k < W_; k += 4) {
      v2f a, b;
      a.x = H[arow][k + ksel]; a.y = H[arow][k + ksel + 1];
      b.x = okc ? Wxi[(k + ksel) * XI_ + col] : 0.0f;
      b.y = okc ? Wxi[(k + ksel + 1) * XI_ + col] : 0.0f;
      acc = wmma4(a, b, acc);
    }
    if (okc)
      for (int i = 0; i < 8; ++i) xi[(rbase + i) * XI_ + col] = acc[i] + bxi[col];
  }
}

// ---------------------------------------------------------------- content addressing
// apply_reset folds memv*(1-r) into the cosine terms (removes the reset pass).
__global__ __launch_bounds__(256)
void k_content(const float* memv, const float* xi, float* out,
               int nh, int key_off, int str_off,
               const int* reset, int t, int apply_reset) {
  int b = blockIdx.x / nh, h = blockIdx.x % nh, tid = threadIdx.x;
  __shared__ float key[W_];
  __shared__ float sim[N_];
  __shared__ float red[256];
  float scale = apply_reset ? (1.0f - (float)reset[b * T_ + t]) : 1.0f;
  for (int i = tid; i < W_; i += 256) key[i] = xi[b * XI_ + key_off + h * W_ + i];
  __syncthreads();
  float p = 0.0f;
  for (int i = tid; i < W_; i += 256) { float v = key[i]; p += v * v; }
  red[tid] = p; __syncthreads();
  for (int s = 128; s > 0; s >>= 1) { if (tid < s) red[tid] += red[tid + s]; __syncthreads(); }
  float kn = sqrtf(red[0]);
  __syncthreads();
  float strength = oneplus(xi[b * XI_ + str_off + h]);
  for (int n = tid; n < N_; n += 256) {
    const float* mr = memv + ((size_t)b * N_ + n) * M_;
    float dot = 0.0f, mn2 = 0.0f;
    for (int m = 0; m < M_; ++m) { float mv = mr[m]; dot += key[m] * mv; mn2 += mv * mv; }
    sim[n] = strength * ((dot * scale) / (kn * (sqrtf(mn2) * scale) + EPS_));
  }
  __syncthreads();
  float mx = -1e30f;
  for (int n = tid; n < N_; n += 256) mx = fmaxf(mx, sim[n]);
  red[tid] = mx; __syncthreads();
  for (int s = 128; s > 0; s >>= 1) { if (tid < s) red[tid] = fmaxf(red[tid], red[tid + s]); __syncthreads(); }
  mx = red[0]; __syncthreads();
  float sum = 0.0f;
  for (int n = tid; n < N_; n += 256) { float e = expf(sim[n] - mx); sim[n] = e; sum += e; }
  red[tid] = sum; __syncthreads();
  for (int s = 128; s > 0; s >>= 1) { if (tid < s) red[tid] += red[tid + s]; __syncthreads(); }
  sum = red[0]; __syncthreads();
  for (int n = tid; n < N_; n += 256) out[((size_t)b * nh + h) * N_ + n] = sim[n] / sum;
}

// ---------------------------------------------------------------- usage / allocation / ww
__global__ __launch_bounds__(256)
void k_usage_alloc(const float* xi, const float* prev_rw, const float* wca,
                   float* usage, float* ww /*in: prev_ww, out: ww*/, float* wsum) {
  int b = blockIdx.x, tid = threadIdx.x;
  __shared__ float u[N_];
  __shared__ float skey[N_];
  __shared__ int sidx[N_];
  __shared__ float cp[N_];
  __shared__ float red[256];
  float fg[R_];
  for (int r = 0; r < R_; ++r) fg[r] = sigm(xi[b * XI_ + 896 + r]);
  for (int n = tid; n < N_; n += 256) {
    float ret = 1.0f;
    for (int r = 0; r < R_; ++r)
      ret *= 1.0f - fg[r] * prev_rw[((size_t)b * R_ + r) * N_ + n];
    float pw = ww[b * N_ + n];
    float uu = usage[b * N_ + n];
    uu = (uu + pw - uu * pw) * ret;
    usage[b * N_ + n] = uu;
    u[n] = uu; skey[n] = uu; sidx[n] = n;
  }
  // stable ascending bitonic argsort (tie-break on index)
  for (int ksz = 2; ksz <= N_; ksz <<= 1)
    for (int j = ksz >> 1; j > 0; j >>= 1) {
      __syncthreads();
      for (int i = tid; i < N_; i += 256) {
        int l = i ^ j;
        if (l > i) {
          float ki = skey[i], kl = skey[l];
          int ii = sidx[i], il = sidx[l];
          bool gt = (ki > kl) || ((ki == kl) && (ii > il));
          bool up = ((i & ksz) == 0);
          if (gt == up) { skey[i] = kl; skey[l] = ki; sidx[i] = il; sidx[l] = ii; }
        }
      }
    }
  __syncthreads();
  // inclusive cumprod over sorted order (Hillis-Steele)
  cp[tid] = skey[tid]; cp[tid + 256] = skey[tid + 256];
  for (int off = 1; off < N_; off <<= 1) {
    __syncthreads();
    float t0 = (tid >= off) ? cp[tid - off] : 1.0f;
    float t1 = cp[tid + 256 - off];
    __syncthreads();
    cp[tid] *= t0; cp[tid + 256] *= t1;
  }
  __syncthreads();
  float wg = sigm(xi[b * XI_ + 918]);   // write gate
  float ag = sigm(xi[b * XI_ + 917]);   // alloc gate
  float acc = 0.0f;
  for (int n = tid; n < N_; n += 256) {
    float alloc = (1.0f - u[n]) * cp[sidx[n]];   // (1-usage)*take_along_axis(cp, si)
    float w = wg * (ag * alloc + (1.0f - ag) * wca[b * N_ + n]);
    ww[b * N_ + n] = w;
    acc += w;
  }
  red[tid] = acc; __syncthreads();
  for (int s = 128; s > 0; s >>= 1) { if (tid < s) red[tid] += red[tid + s]; __syncthreads(); }
  if (tid == 0) wsum[b] = red[0];
}

// ---------------------------------------------------------------- link update (HBM/L2-bound)
__global__ __launch_bounds__(256) void k_link(const float* ww, const float* prec, float* link) {
  size_t idx = (size_t)blockIdx.x * 256 + threadIdx.x;
  int b = (int)(idx / (N_ * N_));
  int rem = (int)(idx % (N_ * N_));
  int i = rem / N_, j = rem % N_;
  float wi = ww[b * N_ + i], wj = ww[b * N_ + j];
  float v = (1.0f - wi - wj) * link[idx] + wi * prec[b * N_ + j];
  link[idx] = (i == j) ? 0.0f : v;
}

__global__ __launch_bounds__(256) void k_prec(const float* ww, const float* wsum, float* prec) {
  unsigned idx = blockIdx.x * 256u + threadIdx.x;   // B*N
  int b = idx / N_;
  prec[idx] = (1.0f - wsum[b]) * prec[idx] + ww[idx];
}

// ---------------------------------------------------------------- memory write (reset folded)
__global__ __launch_bounds__(256)
void k_memwrite(const float* xi, const float* ww, float* memv, const int* reset, int t) {
  size_t idx = (size_t)blockIdx.x * 256 + threadIdx.x;   // B*N*M
  int b = (int)(idx / (N_ * M_));
  int rem = (int)(idx % (N_ * M_));
  int n = rem / M_, m = rem % M_;
  float e = sigm(xi[b * XI_ + 768 + m]);
  float wv = xi[b * XI_ + 640 + m];
  float w = ww[b * N_ + n];
  float scale = 1.0f - (float)reset[b * T_ + t];
  memv[idx] = (memv[idx] * scale) * (1.0f - w * e) + w * wv;
}

// ---------------------------------------------------------------- fw = prev_rw @ link
__global__ __launch_bounds__(256) void k_fw(const float* prev_rw, const float* link, float* fw) {
  int b = blockIdx.x / (N_ / 256);
  int k = (blockIdx.x % (N_ / 256)) * 256 + threadIdx.x;
  const float* rw = prev_rw + (size_t)b * R_ * N_;
  const float* lb = link + (size_t)b * N_ * N_;
  float a0 = 0, a1 = 0, a2 = 0, a3 = 0;
  for (int n = 0; n < N_; ++n) {
    float l = lb[(size_t)n * N_ + k];
    a0 += rw[n] * l; a1 += rw[N_ + n] * l; a2 += rw[2 * N_ + n] * l; a3 += rw[3 * N_ + n] * l;
  }
  fw[((size_t)b * R_ + 0) * N_ + k] = a0;
  fw[((size_t)b * R_ + 1) * N_ + k] = a1;
  fw[((size_t)b * R_ + 2) * N_ + k] = a2;
  fw[((size_t)b * R_ + 3) * N_ + k] = a3;
}

// ---------------------------------------------------------------- bw = prev_rw @ link^T
__global__ __launch_bounds__(128) void k_bw(const float* prev_rw, const float* link, float* bw) {
  int b = blockIdx.x / N_, k = blockIdx.x % N_, tid = threadIdx.x;
  __shared__ float red[4][128];
  const float* lrow = link + ((size_t)b * N_ + k) * N_;
  const float* rw = prev_rw + (size_t)b * R_ * N_;
  float a0 = 0, a1 = 0, a2 = 0, a3 = 0;
  for (int n = tid; n < N_; n += 128) {
    float l = lrow[n];
    a0 += rw[n] * l; a1 += rw[N_ + n] * l; a2 += rw[2 * N_ + n] * l; a3 += rw[3 * N_ + n] * l;
  }
  red[0][tid] = a0; red[1][tid] = a1; red[2][tid] = a2; red[3][tid] = a3;
  __syncthreads();
  for (int s = 64; s > 0; s >>= 1) {
    if (tid < s) for (int r = 0; r < 4; ++r) red[r][tid] += red[r][tid + s];
    __syncthreads();
  }
  if (tid < 4) bw[((size_t)b * R_ + tid) * N_ + k] = red[tid][0];
}

// ---------------------------------------------------------------- rw + reads
__global__ __launch_bounds__(128)
void k_rw_reads(const float* xi, const float* bwv, const float* rca, const float* fwv,
                const float* memv, float* prev_rw, float* prev_reads) {
  int b = blockIdx.x / R_, r = blockIdx.x % R_, tid = threadIdx.x;
  __shared__ float rwl[N_];
  float e0 = xi[b * XI_ + 904 + 3 * r + 0];
  float e1 = xi[b * XI_ + 904 + 3 * r + 1];
  float e2 = xi[b * XI_ + 904 + 3 * r + 2];
  float mx = fmaxf(e0, fmaxf(e1, e2));
  float x0 = expf(e0 - mx), x1 = expf(e1 - mx), x2 = expf(e2 - mx);
  float inv = 1.0f / (x0 + x1 + x2);
  float m0 = x0 * inv, m1 = x1 * inv, m2 = x2 * inv;   // [bw, rca, fw]
  size_t base = ((size_t)b * R_ + r) * N_;
  for (int n = tid; n < N_; n += 128) {
    float v = bwv[base + n] * m0 + rca[base + n] * m1 + fwv[base + n] * m2;
    rwl[n] = v;
    prev_rw[base + n] = v;
  }
  __syncthreads();
  const float* mb = memv + (size_t)b * N_ * M_;
  float acc = 0.0f;
  for (int n = 0; n < N_; ++n) acc += rwl[n] * mb[(size_t)n * M_ + tid];
  prev_reads[(b * R_ + r) * W_ + tid] = acc;
}

// ---------------------------------------------------------------- output layer (WMMA)
__global__ __launch_bounds__(256)
void k_output(const float* hs, const float* reads, const float* Wv, const float* bv,
              const float* Wr, const float* br, float* out, int t) {
  __shared__ float A[B_][640];
  int tid = threadIdx.x;
  for (int i = tid; i < B_ * 640; i += 256) {
    int b = i / 640, j = i % 640;
    A[b][j] = (j < W_) ? hs[b * W_ + j] : reads[b * (R_ * W_) + (j - W_)];
  }
  __syncthreads();
  int wave = tid >> 5, lane = tid & 31;
  int arow = lane & 15, ksel = (lane >> 4) << 1, rbase = (lane >> 4) << 3;
  int col = wave * 16 + arow;
  v8f acc = {};
  for (int k = 0; k < 640; k += 4) {
    v2f a, b;
    a.x = A[arow][k + ksel]; a.y = A[arow][k + ksel + 1];
    int kk = k + ksel;
    b.x = (kk < W_) ? Wv[kk * W_ + col] : Wr[(kk - W_) * W_ + col];
    b.y = (kk + 1 < W_) ? Wv[(kk + 1) * W_ + col] : Wr[(kk + 1 - W_) * W_ + col];
    acc = wmma4(a, b, acc);
  }
  float* ys = out;
  float* hss = out + (size_t)B_ * T_ * W_;
  for (int i = 0; i < 8; ++i) {
    int b = rbase + i;
    ys[(size_t)(b * T_ + t) * W_ + col] = acc[i] + bv[col] + br[col];
  }
  for (int i = tid; i < B_ * W_; i += 256) {
    int b = i / W_, j = i % W_;
    hss[(size_t)(b * T_ + t) * W_ + j] = A[b][j];
  }
}

// ================================================================ host
extern "C" void kernel_launch(void* const* d_in, const int* in_sizes, int n_in,
                              void* d_out, int out_size, void* d_ws, size_t ws_size,
                              hipStream_t stream) {
  const float* X      = (const float*)d_in[0];
  const float* hs_in  = (const float*)d_in[1];
  const float* mem    = (const float*)d_in[2];
  const int*   reset  = (const int*)d_in[3];
  const float* Wb   = (const float*)d_in[4];  const float* bb   = (const float*)d_in[5];
  const float* Wff1 = (const float*)d_in[6];  const float* bff1 = (const float*)d_in[7];
  const float* Wff2 = (const float*)d_in[8];  const float* bff2 = (const float*)d_in[9];
  const float* Wta  = (const float*)d_in[10]; const float* bta  = (const float*)d_in[11];
  const float* Wtb  = (const float*)d_in[12]; const float* btb  = (const float*)d_in[13];
  const float* Wv   = (const float*)d_in[14]; const float* bv   = (const float*)d_in[15];
  const float* Wr   = (const float*)d_in[16]; const float* br   = (const float*)d_in[17];
  const float* Wxi  = (const float*)d_in[18]; const float* bxi  = (const float*)d_in[19];
  float* out = (float*)d_out;
  float* ws = (float*)d_ws;

  float* memv       = ws;                              // 1048576
  float* link       = memv + (size_t)B_ * N_ * M_;     // 4194304
  float* hs         = link + (size_t)B_ * N_ * N_;     // 2048
  float* prev_reads = hs + B_ * W_;                    // 8192
  float* prev_rw    = prev_reads + B_ * R_ * W_;       // 32768
  float* ww         = prev_rw + B_ * R_ * N_;          // 8192 (prev_ww/ww)
  float* usage      = ww + B_ * N_;                    // 8192
  float* prec       = usage + B_ * N_;                 // 8192
  float* xi         = prec + B_ * N_;                  // 16384 (919*16 padded)
  float* wca        = xi + 16384;                      // 8192
  float* rca        = wca + B_ * N_;                   // 32768
  float* fwb        = rca + B_ * R_ * N_;              // 32768
  float* bwb        = fwb + B_ * R_ * N_;              // 32768
  float* wsum       = bwb + B_ * R_ * N_;              // 64
  unsigned zero_n = (unsigned)((wsum + 64) - link);    // zero everything after memv

  k_init_zero<<<(zero_n + 255) / 256, 256, 0, stream>>>(link, zero_n);
  k_init_copy<<<(B_ * N_ * M_) / 256, 256, 0, stream>>>(mem, hs_in, memv, hs);

  for (int t = 0; t < T_; ++t) {
    k_controller<<<1, 256, 0, stream>>>(X, reset, Wb, bb, Wff1, bff1, Wff2, bff2,
                                        Wta, bta, Wtb, btb, Wxi, bxi,
                                        hs, prev_reads, xi, t);
    k_content<<<B_, 256, 0, stream>>>(memv, xi, wca, 1, W_ * R_, 916, reset, t, 1);
    k_usage_alloc<<<B_, 256, 0, stream>>>(xi, prev_rw, wca, usage, ww, wsum);
    k_link<<<(B_ * N_ * N_) / 256, 256, 0, stream>>>(ww, prec, link);
    k_prec<<<(B_ * N_) / 256, 256, 0, stream>>>(ww, wsum, prec);
    k_memwrite<<<(B_ * N_ * M_) / 256, 256, 0, stream>>>(xi, ww, memv, reset, t);
    k_content<<<B_ * R_, 256, 0, stream>>>(memv, xi, rca, R_, 0, 900, reset, t, 0);
    k_fw<<<B_ * (N_ / 256), 256, 0, stream>>>(prev_rw, link, fwb);
    k_bw<<<B_ * N_, 128, 0, stream>>>(prev_rw, link, bwb);
    k_rw_reads<<<B_ * R_, 128, 0, stream>>>(xi, bwb, rca, fwb, memv, prev_rw, prev_reads);
    k_output<<<1, 256, 0, stream>>>(hs, prev_reads, Wv, bv, Wr, br, out, t);
  }
}